// DBASolver_34608846471343
// MI455X (gfx1250) — compile-verified
//
#include <hip/hip_runtime.h>

typedef float v2f __attribute__((ext_vector_type(2)));
typedef float v8f __attribute__((ext_vector_type(8)));

#define B_ 16
#define N_ 196608
#define BLK_PER_BATCH 48
#define WAVES_PER_BLK 8
#define PTS_PER_WAVE (N_ / (BLK_PER_BATCH * WAVES_PER_BLK)) /* 512 */
#define ITERS (PTS_PER_WAVE / 32)                           /* 16  */
#define PT_STRIDE 52                 /* floats per point in LDS (bank-conflict pad) */
#define WAVE_LDS (32 * PT_STRIDE)    /* 1664 floats per wave */

// ---------------------------------------------------------------------------
// Kernel 1: fused Gram/Schur reduction via V_WMMA_F32_16X16X4_F32.
// Per point, 3 K-rows (u0, u1, schur) of an augmented 7-wide row [J, r];
// A-side rows carry the weights (conf / -invHdd), B-side rows are raw, so
// D accumulates  H_pp - term  in [0..5][0..5] and  g_eff  in column 6.
// Two points are packed per K-row (cols 0-6 / 8-14): both diagonal 7x7
// blocks of D are valid partial sums and get added at the end.
// The LDS transpose region is PRIVATE per wave: intra-wave LDS ordering is
// in-order (ISA 7.3), so the hot loop needs no block barriers — only a
// dscnt wait + compiler memory fences.
// ---------------------------------------------------------------------------
__global__ __launch_bounds__(256) void dba_reduce(
    const float* __restrict__ r, const float* __restrict__ w,
    const float* __restrict__ Jp, const float* __restrict__ Jd,
    const float* __restrict__ lmb, float* __restrict__ partial)
{
  __shared__ float lds[WAVES_PER_BLK * WAVE_LDS];
  const int tid  = threadIdx.x;
  const int wid  = tid >> 5;
  const int lane = tid & 31;
  const int b    = blockIdx.y;
  const int blk  = blockIdx.x;
  const long long n0 = (long long)(blk * WAVES_PER_BLK + wid) * PTS_PER_WAVE;
  const float lam = lmb[b];

  float* wlds = lds + wid * WAVE_LDS;

  // Per-lane WMMA operand indices (constant across iterations).
  // A 16x4 f32 layout: lane = (k>=2)*16 + m, vgpr = k&1.  B 4x16 mirrors it.
  const int half = lane >> 4;
  const int mm   = lane & 15;
  const int col  = mm & 7;   // element within a point's row (7 = zero pad)
  const int psel = mm >> 3;  // which point of the packed pair
  int aOff[3][2];
#pragma unroll
  for (int ww = 0; ww < 3; ++ww)
#pragma unroll
    for (int j = 0; j < 2; ++j) {
      int R = 4 * ww + 2 * half + j;      // K-row 0..11
      int pair = R / 3, rt = R - 3 * pair;
      aOff[ww][j] = (pair * 2 + psel) * PT_STRIDE + rt * 8 + col;
    }
  const int sbase = lane * PT_STRIDE;

  v8f acc = {};
  for (int it = 0; it < ITERS; ++it) {
    const size_t pbase = (size_t)b * N_ + (size_t)(n0 + it * 32 + lane);
    const float4* jp4 = (const float4*)(Jp + pbase * 12);
    float4 ja = jp4[0], jb = jp4[1], jc = jp4[2];
    float2 jd = *(const float2*)(Jd + pbase * 2);
    float2 rr = *(const float2*)(r  + pbase * 2);
    float2 wv = *(const float2*)(w  + pbase * 2);
    if (it + 1 < ITERS)                               // global_prefetch_b8
      __builtin_prefetch(Jp + (pbase + 32) * 12, 0, 0);

    const float conf = wv.x, nl = wv.y;
    float u0[7] = {ja.x, ja.y, ja.z, ja.w, jb.x, jb.y, rr.x};
    float u1[7] = {jb.z, jb.w, jc.x, jc.y, jc.z, jc.w, rr.y};
    float h2[7];
#pragma unroll
    for (int i = 0; i < 6; ++i) h2[i] = conf * (u0[i] * jd.x + u1[i] * jd.y); // H_pd
    h2[6] = conf * (jd.x * rr.x + jd.y * rr.y);                               // g_d
    const float hdd = conf * (jd.x * jd.x + jd.y * jd.y);
    const float ih  = 1.0f / fmaxf(hdd + lam + nl, 1e-4f);                    // inv_H_dd

    // A rows at 0,8,16 ; B rows at 24,32,40 (element 7 of each row = 0)
    float4* s4 = (float4*)(wlds + sbase);
    s4[0]  = make_float4(conf*u0[0], conf*u0[1], conf*u0[2], conf*u0[3]);
    s4[1]  = make_float4(conf*u0[4], conf*u0[5], conf*u0[6], 0.f);
    s4[2]  = make_float4(conf*u1[0], conf*u1[1], conf*u1[2], conf*u1[3]);
    s4[3]  = make_float4(conf*u1[4], conf*u1[5], conf*u1[6], 0.f);
    s4[4]  = make_float4(-ih*h2[0], -ih*h2[1], -ih*h2[2], -ih*h2[3]);
    s4[5]  = make_float4(-ih*h2[4], -ih*h2[5], -ih*h2[6], 0.f);
    s4[6]  = make_float4(u0[0], u0[1], u0[2], u0[3]);
    s4[7]  = make_float4(u0[4], u0[5], u0[6], 0.f);
    s4[8]  = make_float4(u1[0], u1[1], u1[2], u1[3]);
    s4[9]  = make_float4(u1[4], u1[5], u1[6], 0.f);
    s4[10] = make_float4(h2[0], h2[1], h2[2], h2[3]);
    s4[11] = make_float4(h2[4], h2[5], h2[6], 0.f);

    // Intra-wave store->load ordering: LDS is in-order per wave; the dscnt
    // wait makes visibility unconditional and the clobber pins the compiler.
    asm volatile("s_wait_dscnt 0x0" ::: "memory");

#pragma unroll
    for (int g = 0; g < 4; ++g) {               // 8-point groups
      const int gb = g * (8 * PT_STRIDE);
#pragma unroll
      for (int ww = 0; ww < 3; ++ww) {          // 3 WMMAs per group
        v2f A, Bv;
        A.x  = wlds[gb + aOff[ww][0]];
        A.y  = wlds[gb + aOff[ww][1]];
        Bv.x = wlds[gb + aOff[ww][0] + 24];
        Bv.y = wlds[gb + aOff[ww][1] + 24];
        acc = __builtin_amdgcn_wmma_f32_16x16x4_f32(
            false, A, false, Bv, (short)0, acc, false, false);
      }
    }
    // WAR fence: next iteration's stores must not be hoisted above these
    // loads at compile time (hardware DS issue is in-order per wave).
    asm volatile("" ::: "memory");
  }

  // Block reduction: D layout = VGPR v: (M=v, N=lane) | (M=v+8, N=lane-16).
  // The red[] staging overlaps other waves' row regions -> block barrier.
  __syncthreads();
  float* red = lds;
#pragma unroll
  for (int v = 0; v < 8; ++v)
    red[(wid * 8 + v) * 32 + lane] = acc[v];
  __syncthreads();
  if (tid < 48) {
    const int mrow = tid >> 3, c = tid & 7;     // c==7 sums zero pads
    float s = 0.f;
    for (int w2 = 0; w2 < WAVES_PER_BLK; ++w2)
      s += red[(w2 * 8 + mrow) * 32 + c]          // block1: (m, c)
         + red[(w2 * 8 + mrow) * 32 + 24 + c];    // block2: (m+8, c+8)
    partial[((size_t)b * BLK_PER_BATCH + blk) * 48 + tid] = s;
  }
}

// ---------------------------------------------------------------------------
// Kernel 2: deterministic partial-sum reduction + per-batch 6x6 solve.
// ---------------------------------------------------------------------------
__global__ void dba_solve(const float* __restrict__ partial,
                          const float* __restrict__ lmb,
                          float* __restrict__ out_pose,
                          float* __restrict__ dp_ws)
{
  __shared__ float sh[48];
  const int b = blockIdx.x, t = threadIdx.x;
  if (t < 48) {
    float s = 0.f;
    for (int blk = 0; blk < BLK_PER_BATCH; ++blk)
      s += partial[((size_t)b * BLK_PER_BATCH + blk) * 48 + t];
    sh[t] = s;
  }
  __syncthreads();
  if (t == 0) {
    const float lam = lmb[b];
    float M[6][7];
    for (int i = 0; i < 6; ++i) {
      for (int j = 0; j < 6; ++j) M[i][j] = sh[i * 8 + j];
      M[i][i] += lam + 0.011f;      // lmbda*I  +  (0.01+0.001)*I
      M[i][6] = sh[i * 8 + 6];      // g_eff
    }
    // Gauss-Jordan with partial pivoting (matches linalg.solve behavior)
    for (int k = 0; k < 6; ++k) {
      int p = k; float best = fabsf(M[k][k]);
      for (int q = k + 1; q < 6; ++q) { float v = fabsf(M[q][k]); if (v > best) { best = v; p = q; } }
      if (p != k) for (int j = 0; j < 7; ++j) { float tmp = M[k][j]; M[k][j] = M[p][j]; M[p][j] = tmp; }
      const float inv = 1.0f / M[k][k];
      for (int j = k; j < 7; ++j) M[k][j] *= inv;
      for (int q = 0; q < 6; ++q) if (q != k) {
        const float f = M[q][k];
        for (int j = k; j < 7; ++j) M[q][j] -= f * M[k][j];
      }
    }
    for (int i = 0; i < 6; ++i) {
      const float dp = M[i][6];
      dp_ws[b * 6 + i]    = dp;                               // unclipped for depth
      out_pose[b * 6 + i] = fminf(fmaxf(dp, -2.0f), 2.0f);    // clipped pose output
    }
  }
}

// ---------------------------------------------------------------------------
// Kernel 3: streaming depth update (bandwidth-bound, B128 loads).
// ---------------------------------------------------------------------------
__global__ __launch_bounds__(256) void dba_depth(
    const float* __restrict__ r, const float* __restrict__ w,
    const float* __restrict__ Jp, const float* __restrict__ Jd,
    const float* __restrict__ lmb, const float* __restrict__ dp_ws,
    float* __restrict__ out_depth)
{
  const size_t idx = (size_t)blockIdx.x * blockDim.x + threadIdx.x;
  if (idx >= (size_t)B_ * N_) return;
  const int b = (int)(idx / N_);
  const float* dpp = dp_ws + b * 6;
  const float4* jp4 = (const float4*)(Jp + idx * 12);
  float4 ja = jp4[0], jb = jp4[1], jc = jp4[2];
  float2 jd = *(const float2*)(Jd + idx * 2);
  float2 rr = *(const float2*)(r  + idx * 2);
  float2 wv = *(const float2*)(w  + idx * 2);
  const float conf = wv.x, nl = wv.y, lam = lmb[b];
  float u0[6] = {ja.x, ja.y, ja.z, ja.w, jb.x, jb.y};
  float u1[6] = {jb.z, jb.w, jc.x, jc.y, jc.z, jc.w};
  float v = 0.f;
#pragma unroll
  for (int i = 0; i < 6; ++i)
    v += conf * (u0[i] * jd.x + u1[i] * jd.y) * dpp[i];       // H_pd . dp
  const float gd  = conf * (jd.x * rr.x + jd.y * rr.y);
  const float hdd = conf * (jd.x * jd.x + jd.y * jd.y);
  const float ih  = 1.0f / fmaxf(hdd + lam + nl, 1e-4f);
  out_depth[idx] = ih * (gd - v);
}

extern "C" void kernel_launch(void* const* d_in, const int* in_sizes, int n_in,
                              void* d_out, int out_size, void* d_ws, size_t ws_size,
                              hipStream_t stream)
{
  const float* r   = (const float*)d_in[0];
  const float* w   = (const float*)d_in[1];
  const float* Jp  = (const float*)d_in[2];
  const float* Jd  = (const float*)d_in[3];
  const float* lmb = (const float*)d_in[4];
  float* out = (float*)d_out;

  float* partial = (float*)d_ws;                         // 16*48*48 floats
  float* dp_ws   = partial + (size_t)B_ * BLK_PER_BATCH * 48;  // 96 floats

  dim3 g1(BLK_PER_BATCH, B_);
  dba_reduce<<<g1, 256, 0, stream>>>(r, w, Jp, Jd, lmb, partial);
  dba_solve<<<B_, 64, 0, stream>>>(partial, lmb, out, dp_ws);
  const size_t tot = (size_t)B_ * N_;
  dba_depth<<<(unsigned)((tot + 255) / 256), 256, 0, stream>>>(
      r, w, Jp, Jd, lmb, dp_ws, out + 96);
}